// DecoderRNN_16484084482923
// MI455X (gfx1250) — compile-verified
//
#include <hip/hip_runtime.h>
#include <hip/hip_bf16.h>
#include <math.h>

#define HIDDEN 2048
#define VOCAB  50257

typedef __attribute__((ext_vector_type(2))) float v2f;
typedef __attribute__((ext_vector_type(8))) float v8f;

// ---------------------------------------------------------------------------
// Kernel 0: x = relu(emb[input_id])  -> ws_x
// ---------------------------------------------------------------------------
__global__ void embed_relu_kernel(const int* __restrict__ ids,
                                  const float* __restrict__ emb,
                                  float* __restrict__ x) {
    const int k = blockIdx.x * blockDim.x + threadIdx.x;
    if (k < HIDDEN) {
        const long long id = (long long)ids[0];
        x[k] = fmaxf(emb[id * (long long)HIDDEN + k], 0.0f);
    }
}

// ---------------------------------------------------------------------------
// Kernel 1: GRU cell. One block (6 waves) per hidden index i.
// wave 0..2: dot(W_ih[i + w*H], x);  wave 3..5: dot(W_hh[i + (w-3)*H], h)
// ---------------------------------------------------------------------------
__global__ void __launch_bounds__(192)
gru_gates_kernel(const float* __restrict__ x,
                 const float* __restrict__ h,
                 const float* __restrict__ W_ih,
                 const float* __restrict__ W_hh,
                 const float* __restrict__ b_ih,
                 const float* __restrict__ b_hh,
                 float* __restrict__ hnew_out,   // d_out + VOCAB
                 float* __restrict__ hnew_ws) {  // workspace copy
    __shared__ float partial[6];

    const int i    = blockIdx.x;          // 0..HIDDEN-1
    const int wave = threadIdx.x >> 5;    // 0..5
    const int lane = threadIdx.x & 31;

    const float* W   = (wave < 3) ? W_ih : W_hh;
    const float* vec = (wave < 3) ? x : h;
    const int gate   = (wave < 3) ? wave : (wave - 3);
    const long long row = (long long)i + (long long)gate * HIDDEN;
    const float* rowp = W + row * HIDDEN;

    float s = 0.0f;
#pragma unroll
    for (int it = 0; it < 16; ++it) {
        const float4 wv = *(const float4*)(rowp + it * 128 + lane * 4);
        const float4 uv = *(const float4*)(vec  + it * 128 + lane * 4);
        s = fmaf(wv.x, uv.x, s);
        s = fmaf(wv.y, uv.y, s);
        s = fmaf(wv.z, uv.z, s);
        s = fmaf(wv.w, uv.w, s);
    }
    // wave32 reduction
#pragma unroll
    for (int off = 16; off > 0; off >>= 1)
        s += __shfl_down(s, off, 32);
    if (lane == 0) partial[wave] = s;
    __syncthreads();

    if (threadIdx.x == 0) {
        const float xr = partial[0] + b_ih[i];
        const float xz = partial[1] + b_ih[i + HIDDEN];
        const float xn = partial[2] + b_ih[i + 2 * HIDDEN];
        const float hr = partial[3] + b_hh[i];
        const float hz = partial[4] + b_hh[i + HIDDEN];
        const float hn = partial[5] + b_hh[i + 2 * HIDDEN];

        const float r = 1.0f / (1.0f + __expf(-(xr + hr)));
        const float z = 1.0f / (1.0f + __expf(-(xz + hz)));
        const float n = tanhf(xn + r * hn);
        const float hv = (1.0f - z) * n + z * h[i];
        hnew_out[i] = hv;
        hnew_ws[i]  = hv;
    }
}

// ---------------------------------------------------------------------------
// Kernel 2: logits = h_new @ W_out^T + b_out via V_WMMA_F32_16X16X4_F32.
// One wave per 16-row tile of W_out. A = h_new broadcast (all 16 M-rows equal),
// B = 4x16 slab of W_out. Accumulate 512 K-steps; D row 0 holds the 16 dots.
// ---------------------------------------------------------------------------
#define NTILE ((VOCAB + 15) / 16)   // 3142

__global__ void __launch_bounds__(256)
logits_wmma_kernel(const float* __restrict__ hnew,
                   const float* __restrict__ W_out,
                   const float* __restrict__ b_out,
                   float* __restrict__ logits) {
    __shared__ float lds_h[HIDDEN];

    // stage h_new into LDS (8 KB), 256 threads x 8 floats
#pragma unroll
    for (int j = 0; j < HIDDEN / 256; ++j)
        lds_h[threadIdx.x + j * 256] = hnew[threadIdx.x + j * 256];
    __syncthreads();

    const int wave = threadIdx.x >> 5;                 // 0..7
    const int tile = blockIdx.x * 8 + wave;            // 16-row tile id
    if (tile >= NTILE) return;                         // whole-wave exit

    const int lane = threadIdx.x & 31;
    const int col  = lane & 15;                        // N index / W_out row in tile
    const int half = lane >> 4;                        // K sub-offset selector
    int row = tile * 16 + col;
    if (row > VOCAB - 1) row = VOCAB - 1;              // clamp (store masked later)
    const float* wrow = W_out + (long long)row * HIDDEN;

    v8f acc = {};
#pragma unroll 4
    for (int k0 = 0; k0 < HIDDEN; k0 += 4) {
        const int ko = k0 + half * 2;
        const v2f a = *(const v2f*)(lds_h + ko);       // A: h broadcast, ds_load_b64
        const v2f b = *(const v2f*)(wrow + ko);        // B: weights, global_load_b64
        acc = __builtin_amdgcn_wmma_f32_16x16x4_f32(
            /*neg_a=*/false, a, /*neg_b=*/false, b,
            /*c_mod=*/(short)0, acc, /*reuse_a=*/false, /*reuse_b=*/false);
    }

    // D row M=0 lives in acc[0] on lanes 0..15
    const int v = tile * 16 + col;
    if (lane < 16 && v < VOCAB)
        logits[v] = acc[0] + b_out[v];
}

// ---------------------------------------------------------------------------
// Kernel 3: log_softmax over 50257 logits. Single 1024-thread block.
// ---------------------------------------------------------------------------
__global__ void __launch_bounds__(1024)
log_softmax_kernel(const float* __restrict__ logits,
                   float* __restrict__ out) {
    __shared__ float red[1024];
    const int tid = threadIdx.x;

    float m = -INFINITY;
    for (int i = tid; i < VOCAB; i += 1024) m = fmaxf(m, logits[i]);
    red[tid] = m;
    __syncthreads();
    for (int s = 512; s > 0; s >>= 1) {
        if (tid < s) red[tid] = fmaxf(red[tid], red[tid + s]);
        __syncthreads();
    }
    const float maxv = red[0];
    __syncthreads();

    float sum = 0.0f;
    for (int i = tid; i < VOCAB; i += 1024) sum += __expf(logits[i] - maxv);
    red[tid] = sum;
    __syncthreads();
    for (int s = 512; s > 0; s >>= 1) {
        if (tid < s) red[tid] += red[tid + s];
        __syncthreads();
    }
    const float lse = maxv + __logf(red[0]);

    for (int i = tid; i < VOCAB; i += 1024) out[i] = logits[i] - lse;
}

// ---------------------------------------------------------------------------
extern "C" void kernel_launch(void* const* d_in, const int* in_sizes, int n_in,
                              void* d_out, int out_size, void* d_ws, size_t ws_size,
                              hipStream_t stream) {
    const int*   ids   = (const int*)  d_in[0];
    const float* h     = (const float*)d_in[1];
    const float* emb   = (const float*)d_in[2];
    const float* W_ih  = (const float*)d_in[3];
    const float* W_hh  = (const float*)d_in[4];
    const float* b_ih  = (const float*)d_in[5];
    const float* b_hh  = (const float*)d_in[6];
    const float* W_out = (const float*)d_in[7];
    const float* b_out = (const float*)d_in[8];

    float* out_ls   = (float*)d_out;            // [0, VOCAB): log_softmax
    float* out_hnew = (float*)d_out + VOCAB;    // [VOCAB, VOCAB+HIDDEN): h_new

    float* ws_x      = (float*)d_ws;            // HIDDEN
    float* ws_hnew   = ws_x + HIDDEN;           // HIDDEN
    float* ws_logits = ws_hnew + HIDDEN;        // VOCAB

    embed_relu_kernel<<<(HIDDEN + 255) / 256, 256, 0, stream>>>(ids, emb, ws_x);

    gru_gates_kernel<<<HIDDEN, 192, 0, stream>>>(ws_x, h, W_ih, W_hh, b_ih, b_hh,
                                                 out_hnew, ws_hnew);

    logits_wmma_kernel<<<(NTILE + 7) / 8, 256, 0, stream>>>(ws_hnew, W_out, b_out,
                                                            ws_logits);

    log_softmax_kernel<<<1, 1024, 0, stream>>>(ws_logits, out_ls);
}